// LogSpacePolynomialCell_29497835388990
// MI455X (gfx1250) — compile-verified
//
#include <hip/hip_runtime.h>
#include <math.h>

#define LOG_ZERO  (-40.0f)
#define SOFT_EPS  (1e-6f)

typedef float v2f __attribute__((ext_vector_type(2)));
typedef float v8f __attribute__((ext_vector_type(8)));
typedef int   v4i __attribute__((ext_vector_type(4)));

typedef __attribute__((address_space(1))) v4i* gvec_ptr;   // global int4*
typedef __attribute__((address_space(3))) v4i* lvec_ptr;   // LDS int4*

#if defined(__HIP_DEVICE_COMPILE__) && __has_builtin(__builtin_amdgcn_global_load_async_to_lds_b128)
#define HAVE_ASYNC_LDS 1
#else
#define HAVE_ASYNC_LDS 0
#endif

__device__ __forceinline__ void wait_async0() {
#if HAVE_ASYNC_LDS
#if __has_builtin(__builtin_amdgcn_s_wait_asynccnt)
    __builtin_amdgcn_s_wait_asynccnt(0);
#else
    asm volatile("s_wait_asynccnt 0x0" ::: "memory");
#endif
#endif
}

// ---------------------------------------------------------------------------
// Out[M,N] = X[M,K] @ W[N,K]^T   using V_WMMA_F32_16X16X4_F32 (full fp32).
// Per-wave tile: 16(M) x 64(N). Block = 8 waves -> 16(M) x 512(N).
// A-tile (16 x 64 of X, shared by all 8 waves) is staged into LDS with
// double-buffered GLOBAL_LOAD_ASYNC_TO_LDS_B128 (s_wait_asynccnt + barrier);
// rows padded to 68 floats so the 16 A-fragment lanes hit distinct bank pairs.
// B fragments are direct global b64 loads (contiguous pair of W row n).
// Fragment layouts per CDNA5 ISA 7.12.2:
//   A 16x4 f32 : lane L holds A[L&15][2*(L>>4) + {0,1}]      -> v2f (from LDS)
//   B 4x16 f32 : lane L holds B[2*(L>>4)+{0,1}][L&15] = W[n][k] pair -> v2f
//   C/D 16x16  : VGPR r: M = r + 8*(L>>4), N = L&15          -> v8f
// fuse_silu_gate: epilogue out = Gate[idx] * silu(acc)  (Gate may alias Out).
// ---------------------------------------------------------------------------
#define TSTRIDE 68   // 64 + 4 floats row padding (bank-conflict free, 16B aligned)

__global__ __launch_bounds__(256)
void wmma_gemm_xwt(const float* __restrict__ X, const float* __restrict__ W,
                   float* __restrict__ Out, const float* __restrict__ Gate,
                   int M, int N, int K, int fuse_silu_gate)
{
    __shared__ float tileA[2][16 * TSTRIDE];

    const int lane = threadIdx.x & 31;
    const int wave = threadIdx.x >> 5;
    const int l15  = lane & 15;
    const int h    = lane >> 4;

    const int m0 = blockIdx.y * 16;
    const int n0 = blockIdx.x * 512 + wave * 64;

    // staging decomposition: 256 threads x float4 = 16 rows x 64 floats
    const int srow = threadIdx.x >> 4;          // 0..15
    const int scol = (threadIdx.x & 15) << 2;   // 0,4,...,60

    v8f acc[4];
    acc[0] = (v8f){}; acc[1] = (v8f){}; acc[2] = (v8f){}; acc[3] = (v8f){};

    const float* wr0 = W + (size_t)(n0 +  0 + l15) * K;
    const float* wr1 = W + (size_t)(n0 + 16 + l15) * K;
    const float* wr2 = W + (size_t)(n0 + 32 + l15) * K;
    const float* wr3 = W + (size_t)(n0 + 48 + l15) * K;
    const float* xsrc = X + (size_t)(m0 + srow) * K + scol;

    auto stage = [&](int buf, int kt) {
        const float* g = xsrc + kt;
        float* l = &tileA[buf][srow * TSTRIDE + scol];
#if HAVE_ASYNC_LDS
        __builtin_amdgcn_global_load_async_to_lds_b128((gvec_ptr)g, (lvec_ptr)l, 0, 0);
#else
        *(float4*)l = *(const float4*)g;
#endif
    };

    const int NT = K / 64;
    stage(0, 0);

    for (int it = 0; it < NT; ++it) {
        const int cur = it & 1;
        const int kt  = it * 64;

        wait_async0();            // my async stores to LDS retired
        __syncthreads();          // whole block's tile visible

        if (it + 1 < NT) {
            stage(cur ^ 1, kt + 64);
            __builtin_prefetch(wr0 + kt + 64, 0, 1);   // global_prefetch_b8
            __builtin_prefetch(wr2 + kt + 64, 0, 1);
        }

        const float* tA = &tileA[cur][(size_t)l15 * TSTRIDE];

        #pragma unroll
        for (int kk = 0; kk < 64; kk += 16) {          // 4 k-steps per chunk
            v2f af[4], b0[4], b1[4], b2[4], b3[4];
            #pragma unroll
            for (int u = 0; u < 4; ++u) {              // issue all loads first
                const int ko = kk + 4 * u + 2 * h;     // even -> 8B aligned
                af[u] = *(const v2f*)(tA  + ko);       // ds_load_b64
                b0[u] = *(const v2f*)(wr0 + kt + ko);
                b1[u] = *(const v2f*)(wr1 + kt + ko);
                b2[u] = *(const v2f*)(wr2 + kt + ko);
                b3[u] = *(const v2f*)(wr3 + kt + ko);
            }
            #pragma unroll
            for (int u = 0; u < 4; ++u) {              // then 16 WMMAs
                acc[0] = __builtin_amdgcn_wmma_f32_16x16x4_f32(false, af[u], false, b0[u], (short)0, acc[0], false, false);
                acc[1] = __builtin_amdgcn_wmma_f32_16x16x4_f32(false, af[u], false, b1[u], (short)0, acc[1], false, false);
                acc[2] = __builtin_amdgcn_wmma_f32_16x16x4_f32(false, af[u], false, b2[u], (short)0, acc[2], false, false);
                acc[3] = __builtin_amdgcn_wmma_f32_16x16x4_f32(false, af[u], false, b3[u], (short)0, acc[3], false, false);
            }
        }
        __syncthreads();          // all reads of 'cur' done before it is restaged
    }

    #pragma unroll
    for (int j = 0; j < 4; ++j) {
        const int ncol = n0 + 16 * j + l15;
        #pragma unroll
        for (int r = 0; r < 8; ++r) {
            const int row = m0 + 8 * h + r;
            const size_t idx = (size_t)row * N + ncol;
            float v = acc[j][r];
            if (fuse_silu_gate) {
                float g = Gate[idx];                       // same thread reads then writes
                Out[idx] = g * (v / (1.0f + expf(-v)));    // compete * silu(w)
            } else {
                Out[idx] = v;
            }
        }
    }
}

// ---------------------------------------------------------------------------
// Scan helpers
// ---------------------------------------------------------------------------
__device__ __forceinline__ float softplusf(float x) {
    return fmaxf(x, 0.0f) + log1pf(expf(-fabsf(x)));
}

__device__ __forceinline__ void signed_log_add(float la, float sa, float lb, float sb,
                                               float& lo, float& so)
{
    const bool a0 = la <= LOG_ZERO + 1.0f;
    const bool b0 = lb <= LOG_ZERO + 1.0f;
    const float mx = fmaxf(la, lb);
    const float mn = fminf(la, lb);
    const float diff = mn - mx;
    const bool  amax = la >= lb;
    const float smax = amax ? sa : sb;
    const float smin = amax ? sb : sa;
    const bool  same = (smax * smin) > 0.0f;
    const float e    = expf(diff);                   // diff <= 0 -> e in (0,1]
    const float lsame = mx + log1pf(e);
    const float lopp  = mx + log1pf(-fminf(e, 0.9999f));
    float lr = same ? lsame : lopp;
    float sr = smax;
    if (a0)        { lr = lb; sr = sb; }
    if (b0 && !a0) { lr = la; sr = sa; }
    lo = lr; so = sr;
}

__device__ __forceinline__ float waveAllMax(float v) {
    #pragma unroll
    for (int o = 16; o > 0; o >>= 1) v = fmaxf(v, __shfl_xor(v, o, 32));
    return v;
}
__device__ __forceinline__ float waveAllSum(float v) {
    #pragma unroll
    for (int o = 16; o > 0; o >>= 1) v += __shfl_xor(v, o, 32);
    return v;
}

__device__ __forceinline__ float blockAllMax(float v, float* sm) {
    const int lane = threadIdx.x & 31, wid = threadIdx.x >> 5, nw = blockDim.x >> 5;
    v = waveAllMax(v);
    if (lane == 0) sm[wid] = v;
    __syncthreads();
    if (wid == 0) {
        float x = (lane < nw) ? sm[lane] : -3.4e38f;
        x = waveAllMax(x);
        if (lane == 0) sm[0] = x;
    }
    __syncthreads();
    float r = sm[0];
    __syncthreads();
    return r;
}
__device__ __forceinline__ float blockAllSum(float v, float* sm) {
    const int lane = threadIdx.x & 31, wid = threadIdx.x >> 5, nw = blockDim.x >> 5;
    v = waveAllSum(v);
    if (lane == 0) sm[wid] = v;
    __syncthreads();
    if (wid == 0) {
        float x = (lane < nw) ? sm[lane] : 0.0f;
        x = waveAllSum(x);
        if (lane == 0) sm[0] = x;
    }
    __syncthreads();
    float r = sm[0];
    __syncthreads();
    return r;
}
// group = 128 contiguous threads = 4 waves (D=1024, N_GROUPS=8)
__device__ __forceinline__ float groupAllMax(float v, float* sm) {
    const int lane = threadIdx.x & 31, wid = threadIdx.x >> 5;
    v = waveAllMax(v);
    if (lane == 0) sm[wid] = v;
    __syncthreads();
    const int g = (wid >> 2) << 2;
    float r = fmaxf(fmaxf(sm[g], sm[g + 1]), fmaxf(sm[g + 2], sm[g + 3]));
    __syncthreads();
    return r;
}
__device__ __forceinline__ float groupAllSum(float v, float* sm) {
    const int lane = threadIdx.x & 31, wid = threadIdx.x >> 5;
    v = waveAllSum(v);
    if (lane == 0) sm[wid] = v;
    __syncthreads();
    const int g = (wid >> 2) << 2;
    float r = sm[g] + sm[g + 1] + sm[g + 2] + sm[g + 3];
    __syncthreads();
    return r;
}

// ---------------------------------------------------------------------------
// Sequential scan: grid = B workgroups (one per batch row / WGP),
// block = D threads (1 thread per feature, 32 wave32 waves). All four GEMMs
// are hoisted out of this kernel, so the T=1024 critical path is pure
// VALU + shfl/LDS reductions.
// ---------------------------------------------------------------------------
__global__ __launch_bounds__(1024)
void scan_kernel(const float* __restrict__ lin_raw,
                 const float* __restrict__ alpha_raw,
                 const float* __restrict__ delta_raw,
                 const float* __restrict__ log_h0, const float* __restrict__ sign_h0,
                 const float* __restrict__ log_r_h, const float* __restrict__ sign_r_h,
                 const float* __restrict__ b_alpha, const float* __restrict__ b_delta,
                 const float* __restrict__ log_gamma,
                 float* __restrict__ out_log, float* __restrict__ out_sign,
                 float* __restrict__ out_compete,   // = output region of d_out (temp)
                 float* __restrict__ hlin_ws,
                 int T, int B, int D)
{
    __shared__ float sm[32];
    const int d = threadIdx.x;
    const int b = blockIdx.x;

    const float logD = logf((float)D);
    const float lrh  = fminf(log_r_h[d], -0.1f);
    const float srh  = sign_r_h[d];
    const float bA   = b_alpha[d];
    const float bD   = b_delta[d];
    const float lg   = log_gamma[d];

    float lh = log_h0[(size_t)b * D + d];
    float sh = sign_h0[(size_t)b * D + d];
    out_log [(size_t)b * D + d] = lh;    // log_h[0]  = log_h0
    out_sign[(size_t)b * D + d] = sh;    // sign_h[0] = sign_h0

    for (int t = 0; t < T; ++t) {
        const size_t base = ((size_t)t * B + b) * D + d;
        const float lin  = lin_raw[base];
        const float araw = alpha_raw[base] + bA;
        const float draw = delta_raw[base] + bD;

        const float alpha = 1.0f + fminf(softplusf(araw), 1.0f);

        // decayed previous state + log-space input
        const float log_rh  = lrh + lh;
        const float sign_rh = srh * sh;
        const float soft_abs = sqrtf(lin * lin + SOFT_EPS * SOFT_EPS);
        const float log_in   = logf(soft_abs);
        const float sign_in  = (lin < 0.0f) ? -1.0f : 1.0f;

        float lv, sv;
        signed_log_add(log_rh, sign_rh, log_in, sign_in, lv, sv);

        const float log_cand  = -softplusf(-(alpha * lv));
        const float log_delta = -softplusf(-draw);       // log_sigmoid(draw)
        const float log_1md   = -softplusf(draw);        // log_sigmoid(-draw)

        float lhn, shn;
        signed_log_add(log_1md + lh, sh, log_delta + log_cand, sv, lhn, shn);

        // fused RMSNorm in log space: logsumexp over D (per batch row)
        const float m   = blockAllMax(2.0f * lhn, sm);
        const float s   = blockAllSum(expf(2.0f * lhn - m), sm);
        const float lse = m + logf(s);
        const float ln  = lhn - 0.5f * (lse - logD) + lg;

        float hlin;
        {
            const bool mask = ln > (LOG_ZERO + 1.0f);
            const float r = shn * expf(fminf(ln, 20.0f));
            hlin = mask ? r : 0.0f;
        }

        // competitive gating: softmax over groups of D/8
        const float gm = groupAllMax(hlin, sm);
        const float e  = expf(hlin - gm);
        const float gs = groupAllSum(e, sm);
        const float compete = e / gs;

        hlin_ws[base]     = hlin;       // GEMM input for hoisted output GEMM
        out_compete[base] = compete;    // consumed (same index) by GEMM epilogue

        const size_t ob = ((size_t)(t + 1) * B + b) * D + d;
        out_log [ob] = lhn;
        out_sign[ob] = shn;
        lh = lhn; sh = shn;
    }
}

// ---------------------------------------------------------------------------
// Launch: 3 input GEMMs -> scan (1 kernel, whole T loop) -> output GEMM.
// Workspace requirement: 4 * T*B*D * 4 bytes (= 128 MiB for T=1024,B=8,D=1024).
// ---------------------------------------------------------------------------
extern "C" void kernel_launch(void* const* d_in, const int* in_sizes, int n_in,
                              void* d_out, int out_size, void* d_ws, size_t ws_size,
                              hipStream_t stream)
{
    const float* x         = (const float*)d_in[0];
    const float* log_h0    = (const float*)d_in[1];
    const float* sign_h0   = (const float*)d_in[2];
    const float* W_x       = (const float*)d_in[3];
    const float* log_r_h   = (const float*)d_in[4];
    const float* sign_r_h  = (const float*)d_in[5];
    const float* W_alpha   = (const float*)d_in[6];
    const float* b_alpha   = (const float*)d_in[7];
    const float* W_delta   = (const float*)d_in[8];
    const float* b_delta   = (const float*)d_in[9];
    const float* W_out     = (const float*)d_in[10];
    const float* log_gamma = (const float*)d_in[11];

    const int D = in_sizes[4];            // log_r_h is [D]
    const int B = in_sizes[1] / D;        // log_h0 is [B,D]
    const int T = in_sizes[0] / (B * D);  // x is [T,B,D]
    const int M = T * B;
    const size_t TBD = (size_t)T * B * D;

    float* ws        = (float*)d_ws;
    float* lin_raw   = ws;
    float* alpha_raw = ws + TBD;
    float* delta_raw = ws + 2 * TBD;
    float* hlin_ws   = ws + 3 * TBD;

    float* out_log  = (float*)d_out;
    float* out_sign = out_log  + (size_t)(T + 1) * B * D;
    float* out_y    = out_sign + (size_t)(T + 1) * B * D;

    const dim3 ggrid(D / 512, M / 16);
    const dim3 gblk(256);

    // Phase A: hoisted input projections (raw GEMM; biases applied in scan)
    wmma_gemm_xwt<<<ggrid, gblk, 0, stream>>>(x, W_x,     lin_raw,   nullptr, M, D, D, 0);
    wmma_gemm_xwt<<<ggrid, gblk, 0, stream>>>(x, W_alpha, alpha_raw, nullptr, M, D, D, 0);
    wmma_gemm_xwt<<<ggrid, gblk, 0, stream>>>(x, W_delta, delta_raw, nullptr, M, D, D, 0);

    // Phase B: the sequential recurrence (one WG per batch row)
    scan_kernel<<<B, D, 0, stream>>>(lin_raw, alpha_raw, delta_raw,
                                     log_h0, sign_h0, log_r_h, sign_r_h,
                                     b_alpha, b_delta, log_gamma,
                                     out_log, out_sign, out_y, hlin_ws, T, B, D);

    // Phase C: hoisted output GEMM; epilogue fuses compete * silu(w_out_h),
    // reading compete from out_y in-place (same thread reads then writes).
    wmma_gemm_xwt<<<ggrid, gblk, 0, stream>>>(hlin_ws, W_out, out_y, out_y, M, D, D, 1);
}